// RADTSE_ForwardOp_25933012533446
// MI455X (gfx1250) — compile-verified
//
#include <hip/hip_runtime.h>

// ---------------------------------------------------------------------------
// RADTSE forward operator for MI455X (gfx1250, wave32, WMMA).
//
// Core math: per echo e, per batch b=(slc,coil):
//   k[b,m] = sum_x Ex[m,x] * ( sum_y C[b,e,x,y] * Ey[m,y] )
// The inner sum is a complex 128x8192x128 fp32 GEMM -> V_WMMA_F32_16X16X4_F32.
// Complex product via 4 real accumulators (rr,ii,ri,ir); F32 WMMA has no A/B
// negation (NEG={CNeg,0,0}), so combine at tile end instead.
// ---------------------------------------------------------------------------

typedef float v2f __attribute__((ext_vector_type(2)));
typedef float v8f __attribute__((ext_vector_type(8)));

#define NSLC 2
#define NPC 3
#define ETL 4
#define NCHA 4
#define NX 128
#define NY 128
#define NLIN 128
#define NCOL 256
#define NLIN_PE (NLIN / ETL)            /* 32   */
#define M_PER_ECHO (NLIN_PE * NCOL)     /* 8192 */
#define NB (NSLC * NCHA)                /* 8    */

// ---------------------------------------------------------------------------
// Kernel 1: coil[b,e,x,y] = (sum_p D[s,e,p] * image[s,p,x,y]) * smaps[s,c,x,y]
// b = s*NCHA + c.  Output flat index matches [b,e,x,y] = thread index.
// ---------------------------------------------------------------------------
__global__ __launch_bounds__(256) void radtse_coil_prep(
    const float* __restrict__ imR, const float* __restrict__ imI,
    const float* __restrict__ smR, const float* __restrict__ smI,
    const float* __restrict__ D,
    float* __restrict__ coilR, float* __restrict__ coilI)
{
    const int idx = blockIdx.x * 256 + threadIdx.x;   // [s,c,e,x,y]
    const int y = idx & 127;
    const int x = (idx >> 7) & 127;
    const int e = (idx >> 14) & 3;
    const int c = (idx >> 16) & 3;
    const int s = idx >> 18;

    const float* dp = D + (s * ETL + e) * NPC;
    const int pix = x * NY + y;
    const int ib = (s * NPC) * NX * NY + pix;

    float gr = 0.f, gi = 0.f;
#pragma unroll
    for (int p = 0; p < NPC; ++p) {
        const float d = dp[p];
        gr += d * imR[ib + p * NX * NY];
        gi += d * imI[ib + p * NX * NY];
    }
    const int sidx = (s * NCHA + c) * NX * NY + pix;
    const float sr = smR[sidx], si = smI[sidx];
    coilR[idx] = gr * sr - gi * si;
    coilI[idx] = gr * si + gi * sr;
}

// ---------------------------------------------------------------------------
// Kernel 2: WMMA NUFFT-style evaluation.
// grid.x = ETL * 256  (256 groups of 32 k-samples per echo)
// block  = 256 threads = 8 waves; wave w -> m-subtile (w>>2)*16, batches
// {2*(w&3), 2*(w&3)+1}.  Phase tables Ey/Ex for the block's 32 samples are
// built once in LDS and reused by every batch and x-tile.
// ---------------------------------------------------------------------------
__global__ __launch_bounds__(256) void radtse_wmma_nufft(
    const float* __restrict__ coilR, const float* __restrict__ coilI,
    const float* __restrict__ traj, const float* __restrict__ mask,
    float* __restrict__ out)
{
    __shared__ float kxs[32], kys[32];
    __shared__ float EyR[128 * 32], EyI[128 * 32];   // [y][m]  (m fastest: bank-clean)
    __shared__ float ExR[128 * 32], ExI[128 * 32];   // [x][m]

    const int t     = threadIdx.x;
    const int e     = blockIdx.x >> 8;       // echo
    const int mgrp  = blockIdx.x & 255;      // 32-sample group within echo
    const int mbase = mgrp * 32;

    if (t < 32) {
        const int mg  = mbase + t;
        const int lin = e * NLIN_PE + (mg >> 8);
        const int col = mg & 255;
        const float* tp = traj + (lin * NCOL + col) * 2;
        kxs[t] = tp[0];
        kys[t] = tp[1];
    }
    __syncthreads();

    // Build exp(-i*k*r) tables: 4096 entries each, 16 per thread.
    for (int i = t; i < 128 * 32; i += 256) {
        const int ml = i & 31;
        const int r  = i >> 5;               // spatial coordinate index
        const float coord = (float)(r - 64); // xs/ys = arange - N/2
        float sy, cy, sx, cx;
        __sincosf(kys[ml] * coord, &sy, &cy);
        __sincosf(kxs[ml] * coord, &sx, &cx);
        EyR[i] = cy;  EyI[i] = -sy;          // exp(-i a) = cos a - i sin a
        ExR[i] = cx;  ExI[i] = -sx;
    }
    __syncthreads();

    const int wave = t >> 5;
    const int lane = t & 31;
    const int mloc = (wave >> 2) * 16;       // 16-m WMMA subtile: 0 or 16
    const int b0   = (wave & 3) * 2;         // first of two batches
    const int hi   = lane >> 4;              // lane half (K / M+8 split)
    const int lm   = lane & 15;              // A row / D column within tile
    const int mcol = mloc + lm;              // m column within block's 32

    for (int bb = 0; bb < 2; ++bb) {
        const int b = b0 + bb;
        const float* cR = coilR + (size_t)(b * ETL + e) * NX * NY;
        const float* cI = coilI + (size_t)(b * ETL + e) * NX * NY;

        float kr = 0.f, ki = 0.f;            // per-lane k-space accumulator

        for (int xt = 0; xt < 8; ++xt) {
            v8f p_rr = {}; v8f p_ii = {}; v8f p_ri = {}; v8f p_ir = {};
            const int xrow = xt * 16 + lm;   // A-matrix row M for this lane

#pragma unroll 4
            for (int yc = 0; yc < 32; ++yc) {
                const int y0 = yc * 4 + 2 * hi;     // A/B: K = 2*hi + v
                // A operand (16x4 f32): lane lm holds row x, K pair y0,y0+1
                v2f ar = *(const v2f*)(cR + xrow * NY + y0);
                v2f ai = *(const v2f*)(cI + xrow * NY + y0);
                // B operand (4x16 f32): VGPR v -> K=2*hi+v, N=lm
                const int i0 = y0 * 32 + mcol;
                const int i1 = i0 + 32;
                v2f br = { EyR[i0], EyR[i1] };
                v2f bi = { EyI[i0], EyI[i1] };

                p_rr = __builtin_amdgcn_wmma_f32_16x16x4_f32(false, ar, false, br, (short)0, p_rr, false, false);
                p_ii = __builtin_amdgcn_wmma_f32_16x16x4_f32(false, ai, false, bi, (short)0, p_ii, false, false);
                p_ri = __builtin_amdgcn_wmma_f32_16x16x4_f32(false, ar, false, bi, (short)0, p_ri, false, false);
                p_ir = __builtin_amdgcn_wmma_f32_16x16x4_f32(false, ai, false, br, (short)0, p_ir, false, false);
            }

            // D layout: lane holds (M = v + 8*hi, N = lm). Apply Ex[m,x] and
            // fold the 8 rows this lane owns into the per-m accumulator.
#pragma unroll
            for (int v = 0; v < 8; ++v) {
                const int xg  = xt * 16 + v + 8 * hi;
                const float tr = p_rr[v] - p_ii[v];
                const float ti = p_ri[v] + p_ir[v];
                const float er = ExR[xg * 32 + mcol];
                const float ei = ExI[xg * 32 + mcol];
                kr += tr * er - ti * ei;
                ki += tr * ei + ti * er;
            }
        }

        // Lane L and L+16 hold complementary x halves of the same m column.
        kr += __shfl_xor(kr, 16, 32);
        ki += __shfl_xor(ki, 16, 32);

        if (hi == 0) {
            const int mg  = mbase + mcol;
            const int lin = e * NLIN_PE + (mg >> 8);
            const int col = mg & 255;
            const int s = b >> 2, c = b & 3;
            const float mv = mask[(s * NLIN + lin) * NCOL + col];
            float* o = out + ((size_t)(((s * NCHA + c) * NLIN + lin) * NCOL + col) << 1);
            o[0] = kr * mv;
            o[1] = ki * mv;
        }
    }
}

// ---------------------------------------------------------------------------
extern "C" void kernel_launch(void* const* d_in, const int* in_sizes, int n_in,
                              void* d_out, int out_size, void* d_ws, size_t ws_size,
                              hipStream_t stream)
{
    const float* imR  = (const float*)d_in[0];
    const float* imI  = (const float*)d_in[1];
    const float* smR  = (const float*)d_in[2];
    const float* smI  = (const float*)d_in[3];
    const float* D    = (const float*)d_in[4];
    const float* mask = (const float*)d_in[5];
    const float* traj = (const float*)d_in[6];
    float* out = (float*)d_out;

    const size_t ncoil = (size_t)NB * ETL * NX * NY;   // 524288 floats
    float* coilR = (float*)d_ws;
    float* coilI = coilR + ncoil;

    radtse_coil_prep<<<(int)(ncoil / 256), 256, 0, stream>>>(
        imR, imI, smR, smI, D, coilR, coilI);

    radtse_wmma_nufft<<<ETL * 256, 256, 0, stream>>>(
        coilR, coilI, traj, mask, out);
}